// OptNet_imp_fun_26147760898259
// MI455X (gfx1250) — compile-verified
//
#include <hip/hip_runtime.h>
#include <hip/hip_bf16.h>
#include <cstddef>

// ---------------------------------------------------------------------------
// Problem constants (from reference): B=128, nz=256, nineq=192, 20 iterations
// ---------------------------------------------------------------------------
#define BB     128
#define NZ     256
#define NQ     192
#define NITER  20
#define EPS_Q  1e-4f

typedef __attribute__((ext_vector_type(2))) float v2f;
typedef __attribute__((ext_vector_type(8))) float v8f;

// ---------------------------------------------------------------------------
// f32 WMMA GEMM:  D = omult * (A[M,K] @ Bt^T + (ADDC ? C : 0))
// Bt is ALWAYS [N,K] row-major (i.e. the transpose of the logical B), so both
// A and B fragments are contiguous 8-byte loads. Symmetric matrices (Q, Qinv)
// are passed directly; G is passed as-is for x@G^T and as a precomputed GT
// for x@G. Compile-time ADDC => branch-free inner loop.
// Each wave computes a 16x32 output strip (two 16x16 accumulators sharing the
// A fragment); 4 waves per workgroup; K advanced 16 per trip (8 WMMAs/trip).
// Layouts per CDNA5 ISA 7.12.2:
//   A  (16x4):  lane(l16,half) holds A[l16, 2*half + j], j=0..1
//   Bt (16x4):  lane(l16,half) holds Bt[l16, 2*half + j] == B[2*half+j, l16]
//   C/D (16x16): vgpr v in lane(l16,half) = C[v + 8*half, l16]
// ---------------------------------------------------------------------------
template<int ADDC>
__global__ void __launch_bounds__(128)
gemm_wmma_f32(const float* __restrict__ A, const float* __restrict__ Bt,
              const float* __restrict__ C, float* __restrict__ D,
              int M, int N, int K, float omult)
{
    const int lane = threadIdx.x & 31;
    const int wave = threadIdx.x >> 5;
    const int half = lane >> 4;
    const int l16  = lane & 15;

    const int tn2  = N >> 5;                    // 32-wide tile columns
    const int tile = blockIdx.x * 4 + wave;
    const int bm   = tile / tn2;
    const int bn   = tile - bm * tn2;
    if (bm >= (M >> 4)) return;

    const int row  = bm * 16 + l16;
    const int col0 = bn * 32 + l16;

    v8f acc0 = {0.f, 0.f, 0.f, 0.f, 0.f, 0.f, 0.f, 0.f};
    v8f acc1 = {0.f, 0.f, 0.f, 0.f, 0.f, 0.f, 0.f, 0.f};

    const float* ap  = A  + (size_t)row  * K + 2 * half;
    const float* bp0 = Bt + (size_t)col0 * K + 2 * half;
    const float* bp1 = bp0 + (size_t)16 * K;

    for (int k0 = 0; k0 < K; k0 += 16) {
        __builtin_prefetch(ap + 16, 0, 3);      // -> global_prefetch_b8
        #pragma unroll
        for (int u = 0; u < 4; ++u) {
            const int kc = 4 * u;
            v2f a  = *(const v2f*)(ap  + kc);
            v2f b0 = *(const v2f*)(bp0 + kc);
            v2f b1 = *(const v2f*)(bp1 + kc);
            acc0 = __builtin_amdgcn_wmma_f32_16x16x4_f32(
                false, a, false, b0, (short)0, acc0, false, false);
            acc1 = __builtin_amdgcn_wmma_f32_16x16x4_f32(
                false, a, false, b1, (short)0, acc1, false, false);
        }
        ap += 16; bp0 += 16; bp1 += 16;
    }

    #pragma unroll
    for (int v = 0; v < 8; ++v) {
        const int m = bm * 16 + v + 8 * half;
        const size_t i0 = (size_t)m * N + col0;
        const size_t i1 = i0 + 16;
        float v0 = acc0[v], v1 = acc1[v];
        if (ADDC) { v0 += C[i0]; v1 += C[i1]; }
        D[i0] = omult * v0;
        D[i1] = omult * v1;
    }
}

// ---------------------------------------------------------------------------
// One-time 256x256 Cholesky, in place in global memory (single workgroup).
// ---------------------------------------------------------------------------
__global__ void __launch_bounds__(256)
chol_inplace_g(float* __restrict__ A, int n)
{
    const int t = threadIdx.x, bd = blockDim.x;
    for (int k = 0; k < n; ++k) {
        if (t == 0) A[(size_t)k * n + k] = sqrtf(A[(size_t)k * n + k]);
        __syncthreads();
        const float lkk = A[(size_t)k * n + k];
        for (int i = k + 1 + t; i < n; i += bd) A[(size_t)i * n + k] /= lkk;
        __syncthreads();
        for (int i = k + 1 + t; i < n; i += bd) {
            const float lik = A[(size_t)i * n + k];
            for (int j = k + 1; j <= i; ++j)
                A[(size_t)i * n + j] -= lik * A[(size_t)j * n + k];
        }
        __syncthreads();
    }
}

// ---------------------------------------------------------------------------
// Solve (L L^T) X = I  column-per-thread  ->  X = Q^{-1}  (one-time).
// ---------------------------------------------------------------------------
__global__ void __launch_bounds__(64)
trisolve_identity(const float* __restrict__ L, float* __restrict__ X, int n)
{
    const int c = blockIdx.x * blockDim.x + threadIdx.x;
    if (c >= n) return;
    for (int i = 0; i < n; ++i) {
        float acc = (i == c) ? 1.f : 0.f;
        for (int j = 0; j < i; ++j)
            acc -= L[(size_t)i * n + j] * X[(size_t)j * n + c];
        X[(size_t)i * n + c] = acc / L[(size_t)i * n + i];
    }
    for (int i = n - 1; i >= 0; --i) {
        float acc = X[(size_t)i * n + c];
        for (int j = i + 1; j < n; ++j)
            acc -= L[(size_t)j * n + i] * X[(size_t)j * n + c];
        X[(size_t)i * n + c] = acc / L[(size_t)i * n + i];
    }
}

// ---------------------------------------------------------------------------
// Batched S = R + diag(s/z) Cholesky, entirely in LDS (148 KB of the 320 KB
// WGP LDS => two workgroups still co-resident per WGP), one workgroup per
// batch element; lower factor spilled to ws (L2-resident, 18.9 MB total).
// ---------------------------------------------------------------------------
#define SLD 193
__global__ void __launch_bounds__(256)
factor_S(const float* __restrict__ R, const float* __restrict__ s,
         const float* __restrict__ z, float* __restrict__ Lc, int unit_d)
{
    extern __shared__ float sm[];           // [NQ][SLD]
    const int b = blockIdx.x, t = threadIdx.x, bd = blockDim.x;
    const int n = NQ;

    for (int idx = t; idx < n * n; idx += bd) {
        const int i = idx / n, j = idx % n;
        float v = R[idx];
        if (i == j) v += unit_d ? 1.0f : (s[b * n + i] / z[b * n + i]);
        sm[i * SLD + j] = v;
    }
    __syncthreads();

    for (int k = 0; k < n; ++k) {
        if (t == 0) sm[k * SLD + k] = sqrtf(sm[k * SLD + k]);
        __syncthreads();
        const float lkk = sm[k * SLD + k];
        for (int i = k + 1 + t; i < n; i += bd) sm[i * SLD + k] /= lkk;
        __syncthreads();
        for (int i = k + 1 + t; i < n; i += bd) {
            const float lik = sm[i * SLD + k];
            for (int j = k + 1; j <= i; ++j)
                sm[i * SLD + j] -= lik * sm[j * SLD + k];
        }
        __syncthreads();
    }

    float* out = Lc + (size_t)b * n * n;
    for (int idx = t; idx < n * n; idx += bd) {
        const int i = idx / n, j = idx % n;
        out[idx] = (j <= i) ? sm[i * SLD + j] : 0.f;
    }
}

// ---------------------------------------------------------------------------
// Batched triangular solve: out = sign * (L L^T)^{-1} rhs, one wave32 / batch.
// ---------------------------------------------------------------------------
__global__ void __launch_bounds__(32)
solve_S(const float* __restrict__ Lc, const float* __restrict__ rhs,
        float* __restrict__ out, float sign)
{
    const int n = NQ;
    const int b = blockIdx.x, lane = threadIdx.x;
    __shared__ float y[NQ];
    const float* L = Lc + (size_t)b * n * n;
    const float* r = rhs + (size_t)b * n;

    for (int i = 0; i < n; ++i) {
        float part = 0.f;
        for (int j = lane; j < i; j += 32) part += L[i * n + j] * y[j];
        for (int off = 16; off; off >>= 1) part += __shfl_down(part, off, 32);
        if (lane == 0) y[i] = (r[i] - part) / L[i * n + i];
        __syncthreads();
    }
    for (int i = n - 1; i >= 0; --i) {
        float part = 0.f;
        for (int j = i + 1 + lane; j < n; j += 32) part += L[j * n + i] * y[j];
        for (int off = 16; off; off >>= 1) part += __shfl_down(part, off, 32);
        if (lane == 0) y[i] = (y[i] - part) / L[i * n + i];
        __syncthreads();
    }
    for (int i = lane; i < n; i += 32) out[(size_t)b * n + i] = sign * y[i];
}

// ---------------------------------------------------------------------------
// Small elementwise / reduction kernels
// ---------------------------------------------------------------------------
__global__ void k_ewmul(const float* a, const float* b, float* o, int n) {
    int i = blockIdx.x * blockDim.x + threadIdx.x;
    if (i < n) o[i] = a[i] * b[i];
}
__global__ void k_diag_add(float* A, int n, float eps) {
    int i = blockIdx.x * blockDim.x + threadIdx.x;
    if (i < n) A[(size_t)i * n + i] += eps;
}
__global__ void k_transpose_g(const float* __restrict__ G, float* __restrict__ GT) {
    int i = blockIdx.x * blockDim.x + threadIdx.x;      // over NZ*NQ
    if (i < NZ * NQ) {
        int nn = i / NQ, kk = i - nn * NQ;
        GT[i] = G[(size_t)kk * NZ + nn];                // GT[n][k] = G[k][n]
    }
}
__global__ void k_h(const float* __restrict__ G, const float* __restrict__ s0,
                    const float* __restrict__ z0, float* __restrict__ h) {
    int j = blockIdx.x * blockDim.x + threadIdx.x;
    if (j < NQ) {
        float acc = s0[j];
        for (int k = 0; k < NZ; ++k) acc += G[(size_t)j * NZ + k] * z0[k];
        h[j] = acc;
    }
}
__global__ void k_bcast_row(const float* h, float* o, int n) {   // o[b,j] = h[j]
    int i = blockIdx.x * blockDim.x + threadIdx.x;
    if (i < n) o[i] = h[i % NQ];
}
__global__ void k_sub_rowb(const float* s, const float* h, float* o, int n) { // o = s - h[j]
    int i = blockIdx.x * blockDim.x + threadIdx.x;
    if (i < n) o[i] = s[i] - h[i % NQ];
}
__global__ void k_sub(const float* a, const float* b, float* o, int n) {
    int i = blockIdx.x * blockDim.x + threadIdx.x;
    if (i < n) o[i] = a[i] - b[i];
}
__global__ void k_dsa(const float* s, const float* z, const float* w, float* o, int n) {
    int i = blockIdx.x * blockDim.x + threadIdx.x;
    if (i < n) o[i] = -(z[i] + w[i]) * s[i] / z[i];
}
__global__ void k_dsc(const float* s, const float* z, const float* rsc,
                      const float* w2, float* o, int n) {
    int i = blockIdx.x * blockDim.x + threadIdx.x;
    if (i < n) o[i] = (-rsc[i] - w2[i]) * s[i] / z[i];
}
__global__ void k_init_sz(const float* w, float* s, float* z, int n) {
    int i = blockIdx.x * blockDim.x + threadIdx.x;
    if (i < n) { s[i] = -w[i]; z[i] = w[i]; }
}
__global__ void k_sum2(const float* a, const float* b, const float* c,
                       const float* d, float* o1, float* o2, int n) {
    int i = blockIdx.x * blockDim.x + threadIdx.x;
    if (i < n) { o1[i] = a[i] + b[i]; o2[i] = c[i] + d[i]; }
}
__global__ void k_upd_x(float* x, const float* dxa, const float* dxc,
                        const float* alpha, int n) {
    int i = blockIdx.x * blockDim.x + threadIdx.x;
    if (i < n) x[i] += alpha[i / NZ] * (dxa[i] + dxc[i]);
}
__global__ void k_upd_sz(float* s, float* z, const float* dst, const float* dzt,
                         const float* alpha, int n) {
    int i = blockIdx.x * blockDim.x + threadIdx.x;
    if (i < n) { float a = alpha[i / NQ]; s[i] += a * dst[i]; z[i] += a * dzt[i]; }
}
// row-min shift:  s = (min(s) < 0) ? s - min(s) + 1 : s   (one block per row)
__global__ void __launch_bounds__(64)
k_shift_pos(float* v)
{
    __shared__ float red[64];
    const int b = blockIdx.x, t = threadIdx.x;
    float m = 1e30f;
    for (int j = t; j < NQ; j += 64) m = fminf(m, v[b * NQ + j]);
    red[t] = m; __syncthreads();
    for (int st = 32; st; st >>= 1) {
        if (t < st) red[t] = fminf(red[t], red[t + st]);
        __syncthreads();
    }
    const float ms = red[0];
    if (ms < 0.f)
        for (int j = t; j < NQ; j += 64) v[b * NQ + j] += 1.f - ms;
}
// alpha[b] = min(mult * min_j step, 1), step over (v1,dv1) and (v2,dv2)
__global__ void __launch_bounds__(64)
k_step_alpha(const float* v1, const float* dv1, const float* v2,
             const float* dv2, float* alpha, float mult)
{
    __shared__ float red[64];
    const int b = blockIdx.x, t = threadIdx.x;
    float m = 1e30f;
    for (int j = t; j < NQ; j += 64) {
        const float d1 = dv1[b * NQ + j];
        if (d1 < 0.f) m = fminf(m, -v1[b * NQ + j] / d1);
        const float d2 = dv2[b * NQ + j];
        if (d2 < 0.f) m = fminf(m, -v2[b * NQ + j] / d2);
    }
    red[t] = m; __syncthreads();
    for (int st = 32; st; st >>= 1) {
        if (t < st) red[t] = fminf(red[t], red[t + st]);
        __syncthreads();
    }
    if (t == 0) alpha[b] = fminf(mult * red[0], 1.0f);
}
// corrector RHS: mu, sigma, rs_c and hv2 = rs_c * s / z  (one block per row)
__global__ void __launch_bounds__(64)
k_corr(const float* s, const float* z, const float* dsa, const float* w,
       const float* alpha, float* rsc, float* hv2)
{
    __shared__ float r1[64], r2[64];
    const int b = blockIdx.x, t = threadIdx.x;
    const float a = alpha[b];
    float s1 = 0.f, s2 = 0.f;
    for (int j = t; j < NQ; j += 64) {
        const float sv = s[b * NQ + j], zv = z[b * NQ + j];
        s1 += sv * zv;
        s2 += (sv + a * dsa[b * NQ + j]) * (zv + a * w[b * NQ + j]);
    }
    r1[t] = s1; r2[t] = s2; __syncthreads();
    for (int st = 32; st; st >>= 1) {
        if (t < st) { r1[t] += r1[t + st]; r2[t] += r2[t + st]; }
        __syncthreads();
    }
    const float ssz = r1[0];
    const float mu  = fabsf(ssz / (float)NQ);
    const float rat = r2[0] / ssz;
    const float sig = rat * rat * rat;
    for (int j = t; j < NQ; j += 64) {
        const float sv = s[b * NQ + j], zv = z[b * NQ + j];
        const float rc = (-mu * sig + dsa[b * NQ + j] * w[b * NQ + j]) / sv;
        rsc[b * NQ + j] = rc;
        hv2[b * NQ + j] = rc * sv / zv;
    }
}

// ---------------------------------------------------------------------------
// Workspace layout (float offsets)
// ---------------------------------------------------------------------------
namespace {
constexpr size_t O_LEFF = 0;
constexpr size_t O_Q    = O_LEFF + (size_t)NZ * NZ;
constexpr size_t O_LQ   = O_Q    + (size_t)NZ * NZ;
constexpr size_t O_QINV = O_LQ   + (size_t)NZ * NZ;
constexpr size_t O_TGQ  = O_QINV + (size_t)NZ * NZ;
constexpr size_t O_GT   = O_TGQ  + (size_t)NQ * NZ;
constexpr size_t O_R    = O_GT   + (size_t)NZ * NQ;
constexpr size_t O_H    = O_R    + (size_t)NQ * NQ;
constexpr size_t O_S    = O_H    + 256;
constexpr size_t O_Z    = O_S    + (size_t)BB * NQ;
constexpr size_t O_RX   = O_Z    + (size_t)BB * NQ;
constexpr size_t O_RZ   = O_RX   + (size_t)BB * NZ;
constexpr size_t O_T1   = O_RZ   + (size_t)BB * NQ;
constexpr size_t O_HV   = O_T1   + (size_t)BB * NZ;
constexpr size_t O_W    = O_HV   + (size_t)BB * NQ;
constexpr size_t O_T2   = O_W    + (size_t)BB * NQ;
constexpr size_t O_DXA  = O_T2   + (size_t)BB * NZ;
constexpr size_t O_DSA  = O_DXA  + (size_t)BB * NZ;
constexpr size_t O_RSC  = O_DSA  + (size_t)BB * NQ;
constexpr size_t O_HV2  = O_RSC  + (size_t)BB * NQ;
constexpr size_t O_W2   = O_HV2  + (size_t)BB * NQ;
constexpr size_t O_T3   = O_W2   + (size_t)BB * NQ;
constexpr size_t O_DXC  = O_T3   + (size_t)BB * NZ;
constexpr size_t O_DSC  = O_DXC  + (size_t)BB * NZ;
constexpr size_t O_DST  = O_DSC  + (size_t)BB * NQ;
constexpr size_t O_DZT  = O_DST  + (size_t)BB * NQ;
constexpr size_t O_AL   = O_DZT  + (size_t)BB * NQ;
constexpr size_t O_AL2  = O_AL   + 128;
constexpr size_t O_LC   = O_AL2  + 128;

// D = omult * (A[M,K] @ Bt^T + C);  Bt is [N,K] row-major
inline void gemm(const float* A, const float* Bt, const float* C, float* D,
                 int M, int N, int K, float omult, hipStream_t st)
{
    const int tiles = (M >> 4) * (N >> 5);      // one wave per 16x32 strip
    dim3 g((tiles + 3) >> 2);
    dim3 blk(128);
    if (C) gemm_wmma_f32<1><<<g, blk, 0, st>>>(A, Bt, C, D, M, N, K, omult);
    else   gemm_wmma_f32<0><<<g, blk, 0, st>>>(A, Bt, C, D, M, N, K, omult);
}
inline dim3 gs(int n) { return dim3((n + 255) / 256); }
} // namespace

// ---------------------------------------------------------------------------
extern "C" void kernel_launch(void* const* d_in, const int* in_sizes, int n_in,
                              void* d_out, int out_size, void* d_ws, size_t ws_size,
                              hipStream_t stream)
{
    (void)in_sizes; (void)n_in; (void)out_size; (void)ws_size;

    const float* p  = (const float*)d_in[0];   // [B,NZ]  (reference arg 'x' == linear term)
    const float* Mm = (const float*)d_in[1];   // [NZ,NZ]
    const float* Lm = (const float*)d_in[2];   // [NZ,NZ]
    const float* G  = (const float*)d_in[3];   // [NQ,NZ]
    const float* s0 = (const float*)d_in[4];   // [NQ]
    const float* z0 = (const float*)d_in[5];   // [NZ]

    float* ws   = (float*)d_ws;
    float* Leff = ws + O_LEFF;  float* Q    = ws + O_Q;    float* Lq   = ws + O_LQ;
    float* Qinv = ws + O_QINV;  float* TGQ  = ws + O_TGQ;  float* GT   = ws + O_GT;
    float* R    = ws + O_R;     float* h    = ws + O_H;    float* s    = ws + O_S;
    float* z    = ws + O_Z;     float* rx   = ws + O_RX;   float* rz   = ws + O_RZ;
    float* t1   = ws + O_T1;    float* hv   = ws + O_HV;   float* w    = ws + O_W;
    float* t2   = ws + O_T2;    float* dxa  = ws + O_DXA;  float* dsa  = ws + O_DSA;
    float* rsc  = ws + O_RSC;   float* hv2  = ws + O_HV2;  float* w2   = ws + O_W2;
    float* t3   = ws + O_T3;    float* dxc  = ws + O_DXC;  float* dsc  = ws + O_DSC;
    float* dst  = ws + O_DST;   float* dzt  = ws + O_DZT;  float* al   = ws + O_AL;
    float* al2  = ws + O_AL2;   float* Lc   = ws + O_LC;
    float* x    = (float*)d_out;               // state x lives directly in d_out

    const int nQQ = NZ * NZ, nBZ = BB * NZ, nBQ = BB * NQ;
    const size_t sldsBytes = (size_t)NQ * SLD * sizeof(float);

    // ---- one-time setup --------------------------------------------------
    k_ewmul<<<gs(nQQ), 256, 0, stream>>>(Mm, Lm, Leff, nQQ);
    gemm(Leff, Leff, nullptr, Q, NZ, NZ, NZ, 1.f, stream);         // Q = Leff Leff^T (symmetric)
    k_diag_add<<<gs(NZ), 256, 0, stream>>>(Q, NZ, EPS_Q);
    hipMemcpyAsync(Lq, Q, (size_t)nQQ * sizeof(float), hipMemcpyDeviceToDevice, stream);
    chol_inplace_g<<<1, 256, 0, stream>>>(Lq, NZ);
    trisolve_identity<<<4, 64, 0, stream>>>(Lq, Qinv, NZ);         // Qinv = Q^{-1} (symmetric)
    k_transpose_g<<<gs(NZ * NQ), 256, 0, stream>>>(G, GT);         // GT[n][k] = G[k][n]
    gemm(G, Qinv, nullptr, TGQ, NQ, NZ, NZ, 1.f, stream);          // TGQ = G Qinv   (Bt=Qinv sym)
    gemm(TGQ, G, nullptr, R, NQ, NQ, NZ, 1.f, stream);             // R = TGQ G^T    (Bt=G)
    k_h<<<gs(NQ), 256, 0, stream>>>(G, s0, z0, h);

    // ---- initial (x,s,z): solve_kkt(d=1, rx=p, rs=0, rz=-h) --------------
    gemm(p, Qinv, nullptr, t1, BB, NZ, NZ, 1.f, stream);           // t1 = p Qinv
    k_bcast_row<<<gs(nBQ), 256, 0, stream>>>(h, hv, nBQ);          // hv = h
    gemm(t1, G, hv, hv, BB, NQ, NZ, 1.f, stream);                  // hv += t1 G^T
    factor_S<<<BB, 256, sldsBytes, stream>>>(R, s, z, Lc, 1);      // chol(R+I)
    solve_S<<<BB, 32, 0, stream>>>(Lc, hv, w, -1.f);               // w = -S\hv
    gemm(w, GT, p, t2, BB, NZ, NQ, 1.f, stream);                   // t2 = wG + p   (Bt=GT)
    gemm(t2, Qinv, nullptr, x, BB, NZ, NZ, -1.f, stream);          // x = -(t2 Qinv)
    k_init_sz<<<gs(nBQ), 256, 0, stream>>>(w, s, z, nBQ);          // s=-w, z=w
    k_shift_pos<<<BB, 64, 0, stream>>>(s);
    k_shift_pos<<<BB, 64, 0, stream>>>(z);

    // ---- PDIPM iterations ------------------------------------------------
    for (int it = 0; it < NITER; ++it) {
        // residuals
        gemm(x, Q, p, rx, BB, NZ, NZ, 1.f, stream);                // rx = xQ + p   (Bt=Q sym)
        gemm(z, GT, rx, rx, BB, NZ, NQ, 1.f, stream);              // rx += zG      (Bt=GT)
        k_sub_rowb<<<gs(nBQ), 256, 0, stream>>>(s, h, rz, nBQ);    // rz = s - h
        gemm(x, G, rz, rz, BB, NQ, NZ, 1.f, stream);               // rz += xG^T    (Bt=G)
        // predictor
        gemm(rx, Qinv, nullptr, t1, BB, NZ, NZ, 1.f, stream);      // t1 = rx Qinv
        k_sub<<<gs(nBQ), 256, 0, stream>>>(s, rz, hv, nBQ);        // hv = s - rz   (rs/d == s)
        gemm(t1, G, hv, hv, BB, NQ, NZ, 1.f, stream);              // hv += t1 G^T
        factor_S<<<BB, 256, sldsBytes, stream>>>(R, s, z, Lc, 0);  // chol(R+diag(s/z))
        solve_S<<<BB, 32, 0, stream>>>(Lc, hv, w, -1.f);           // w = -S\hv  (dz_a)
        gemm(w, GT, rx, t2, BB, NZ, NQ, 1.f, stream);              // t2 = wG + rx
        gemm(t2, Qinv, nullptr, dxa, BB, NZ, NZ, -1.f, stream);    // dx_a = -(t2 Qinv)
        k_dsa<<<gs(nBQ), 256, 0, stream>>>(s, z, w, dsa, nBQ);     // ds_a = -(z+w)s/z
        k_step_alpha<<<BB, 64, 0, stream>>>(z, w, s, dsa, al, 1.0f);
        // corrector
        k_corr<<<BB, 64, 0, stream>>>(s, z, dsa, w, al, rsc, hv2); // rs_c, hv2=rs_c*s/z
        solve_S<<<BB, 32, 0, stream>>>(Lc, hv2, w2, -1.f);         // w2 = -S\hv2 (dz_c)
        gemm(w2, GT, nullptr, t3, BB, NZ, NQ, 1.f, stream);        // t3 = w2 G
        gemm(t3, Qinv, nullptr, dxc, BB, NZ, NZ, -1.f, stream);    // dx_c = -(t3 Qinv)
        k_dsc<<<gs(nBQ), 256, 0, stream>>>(s, z, rsc, w2, dsc, nBQ); // ds_c
        // combine + step
        k_sum2<<<gs(nBQ), 256, 0, stream>>>(dsa, dsc, w, w2, dst, dzt, nBQ);
        k_step_alpha<<<BB, 64, 0, stream>>>(z, dzt, s, dst, al2, 0.999f);
        k_upd_x<<<gs(nBZ), 256, 0, stream>>>(x, dxa, dxc, al2, nBZ);
        k_upd_sz<<<gs(nBQ), 256, 0, stream>>>(s, z, dst, dzt, al2, nBQ);
    }
}